// RepetitionPenaltyLoss_28759101014353
// MI455X (gfx1250) — compile-verified
//
#include <hip/hip_runtime.h>
#include <hip/hip_bf16.h>

typedef float v4f __attribute__((ext_vector_type(4)));

// Problem constants from the reference setup.
#define WINDOW_W   10
#define WEIGHT_F   0.2f
#define B_DIM      2
#define S_DIM      2048
#define V_DIM      32000
#define TW_DIM     (S_DIM - WINDOW_W)        // 2038
#define NROWS      (B_DIM * TW_DIM)          // 4076
#define THREADS    256                        // 8 wave32s per block
#define NWAVES     (THREADS / 32)
#define STAGE_F    (THREADS * 4)              // 1024 floats = 4096 B per stage
#define STAGE_B    (STAGE_F * 4)
#define NBUF       4                          // 4-deep async ring (16 KB LDS)
#define NFULL      (V_DIM / STAGE_F)          // 31 full async stages
#define TAIL_F     (V_DIM - NFULL * STAGE_F)  // 256 tail floats
#define TAIL_V4    (TAIL_F / 4)               // 64 float4 tail loads

// ---- CDNA5 async global->LDS copy (ASYNCcnt-tracked, no VGPR return) ----
__device__ __forceinline__ void async_copy_b128(unsigned lds_byte_addr,
                                                unsigned global_byte_off,
                                                const float* base) {
  // saddr form: VDST = per-lane LDS byte address, VADDR = 32-bit byte offset,
  // SADDR = uniform 64-bit base. Stream is read-once -> non-temporal.
  asm volatile("global_load_async_to_lds_b128 %0, %1, %2 th:TH_LOAD_NT"
               :
               : "v"(lds_byte_addr), "v"(global_byte_off), "s"(base)
               : "memory");
}
#define ASYNC_WAIT(imm) asm volatile("s_wait_asynccnt " #imm ::: "memory")

__global__ __launch_bounds__(THREADS)
void rp_row_softmax_gather(const float* __restrict__ logits,
                           const int* __restrict__ input_ids,
                           float* __restrict__ partial) {
  __shared__ float tile[NBUF][STAGE_F];   // 16 KB ring buffer
  __shared__ float red_m[NWAVES];
  __shared__ float red_s[NWAVES];

  const int row = blockIdx.x;                 // 0 .. NROWS-1
  const int b   = row / TW_DIM;
  const int t   = row - b * TW_DIM;
  const int tid = threadIdx.x;

  const float* rowp = logits + ((size_t)b * S_DIM + (size_t)(WINDOW_W + t)) * V_DIM;

  // Low 32 bits of the flat address of a __shared__ object are its LDS offset.
  const unsigned lds0 = (unsigned)(size_t)(&tile[0][0]) + (unsigned)(tid * 16);

  // Prologue: keep 3 stages in flight.
  async_copy_b128(lds0 + 0u * STAGE_B, 0u * STAGE_B + (unsigned)(tid * 16), rowp);
  async_copy_b128(lds0 + 1u * STAGE_B, 1u * STAGE_B + (unsigned)(tid * 16), rowp);
  async_copy_b128(lds0 + 2u * STAGE_B, 2u * STAGE_B + (unsigned)(tid * 16), rowp);

  float m = -3.4e38f;
  float s = 0.0f;

  // Each thread consumes exactly the 16 bytes it loaded: no barriers needed,
  // only the per-wave ASYNCcnt wait (async loads complete in order).
  auto consume = [&](int k) {
    const v4f v = *reinterpret_cast<const v4f*>(&tile[k & (NBUF - 1)][tid * 4]);
    const float lm = fmaxf(fmaxf(v.x, v.y), fmaxf(v.z, v.w));
    if (lm > m) { s *= __expf(m - lm); m = lm; }
    s += __expf(v.x - m) + __expf(v.y - m) + __expf(v.z - m) + __expf(v.w - m);
  };

  for (int k = 0; k < NFULL - 3; ++k) {
    async_copy_b128(lds0 + (unsigned)(((k + 3) & (NBUF - 1)) * STAGE_B),
                    (unsigned)((k + 3) * STAGE_B + tid * 16), rowp);
    ASYNC_WAIT(0x3);
    consume(k);
  }
  ASYNC_WAIT(0x2); consume(NFULL - 3);
  ASYNC_WAIT(0x1); consume(NFULL - 2);
  ASYNC_WAIT(0x0); consume(NFULL - 1);

  // 256-float tail: direct non-temporal vector loads by the first 64 threads.
  if (tid < TAIL_V4) {
    const v4f v = __builtin_nontemporal_load(
        reinterpret_cast<const v4f*>(rowp + NFULL * STAGE_F + tid * 4));
    const float lm = fmaxf(fmaxf(v.x, v.y), fmaxf(v.z, v.w));
    if (lm > m) { s *= __expf(m - lm); m = lm; }
    s += __expf(v.x - m) + __expf(v.y - m) + __expf(v.z - m) + __expf(v.w - m);
  }

  // Intra-wave (max, sumexp) butterfly merge (wave32).
  for (int off = 16; off > 0; off >>= 1) {
    const float m2 = __shfl_xor(m, off, 32);
    const float s2 = __shfl_xor(s, off, 32);
    const float nm = fmaxf(m, m2);
    s = s * __expf(m - nm) + s2 * __expf(m2 - nm);
    m = nm;
  }
  if ((tid & 31) == 0) { red_m[tid >> 5] = m; red_s[tid >> 5] = s; }
  __syncthreads();

  if (tid == 0) {
    float M = red_m[0], S = red_s[0];
#pragma unroll
    for (int w = 1; w < NWAVES; ++w) {
      const float m2 = red_m[w], s2 = red_s[w];
      const float nm = fmaxf(M, m2);
      S = S * __expf(M - nm) + s2 * __expf(m2 - nm);
      M = nm;
    }

    const int* ids = input_ids + (size_t)b * S_DIM + t;   // window tokens t..t+9
    int tok[WINDOW_W];
#pragma unroll
    for (int j = 0; j < WINDOW_W; ++j) tok[j] = ids[j];
    float g[WINDOW_W];
#pragma unroll
    for (int j = 0; j < WINDOW_W; ++j) g[j] = rowp[tok[j]];

    float acc = 0.0f;
#pragma unroll
    for (int j = 0; j < WINDOW_W; ++j) {
      bool dup = false;
#pragma unroll
      for (int kk = 0; kk < WINDOW_W; ++kk)
        if (kk < j) dup = dup || (tok[kk] == tok[j]);
      acc += dup ? 0.0f : __expf(g[j] - M);
    }
    partial[row] = acc / S;
  }
}

__global__ __launch_bounds__(THREADS)
void rp_finalize(const float* __restrict__ partial, float* __restrict__ out) {
  __shared__ float red[THREADS];
  float acc = 0.0f;
  for (int i = threadIdx.x; i < NROWS; i += THREADS) acc += partial[i];
  red[threadIdx.x] = acc;
  __syncthreads();
  for (int off = THREADS / 2; off > 0; off >>= 1) {
    if (threadIdx.x < off) red[threadIdx.x] += red[threadIdx.x + off];
    __syncthreads();
  }
  if (threadIdx.x == 0)
    out[0] = red[0] * (WEIGHT_F / (float)(B_DIM * TW_DIM));
}

extern "C" void kernel_launch(void* const* d_in, const int* in_sizes, int n_in,
                              void* d_out, int out_size, void* d_ws, size_t ws_size,
                              hipStream_t stream) {
  (void)in_sizes; (void)n_in; (void)out_size; (void)ws_size;
  const float* logits    = (const float*)d_in[0];
  const int*   input_ids = (const int*)d_in[1];
  float*       out       = (float*)d_out;
  float*       partial   = (float*)d_ws;     // NROWS floats of scratch

  rp_row_softmax_gather<<<NROWS, THREADS, 0, stream>>>(logits, input_ids, partial);
  rp_finalize<<<1, THREADS, 0, stream>>>(partial, out);
}